// ODEfuncHNN_31044023615617
// MI455X (gfx1250) — compile-verified
//
#include <hip/hip_runtime.h>

// out[i] = ( dH/dy2 , -dH/dy1 )  for H(y) = sum_k C_k * y1^e1k * y2^e2k,
// EXPS ordered by total degree d asc, then e1 desc  =>  idx(e1,e2) = d(d+1)/2 + e2.
//
// Both gradient components are degree-4 polynomials over the 15 monomials
// y1^a*y2^b (a+b<=4), padded to 16.  Per 16-point tile:
//     out_tile[16x2] = M[16x16] @ W[16x2]   via 4 x v_wmma_f32_16x16x4_f32.
// W col0 = (b+1)*C[idx(a,b+1)]  (dH/dy2);  col1 = -(a+1)*C[idx(a+1,b)] (-dH/dy1).

typedef float v2f __attribute__((ext_vector_type(2)));
typedef float v8f __attribute__((ext_vector_type(8)));

// Monomials + 4-chunk WMMA accumulation for one 16-point tile.
__device__ __forceinline__
v8f hnn_compute(v2f yv, bool hi, v2f b0, v2f b1, v2f b2, v2f b3)
{
    const float y1 = yv.x, z1 = yv.y;

    const float x2 = y1*y1, x3 = x2*y1, x4 = x2*x2;
    const float z2 = z1*z1, z3 = z2*z1, z4 = z2*z2;
    const float m4 = y1*z1, m7 = x2*z1, m8 = y1*z2;
    const float m11 = x3*z1, m12 = x2*z2, m13 = y1*z3;

    // A 16x4 chunks: lane holds K-pair {0,1} (lo half) or {2,3} (hi half) of its row.
    const v2f a0 = { hi ? z1 : 1.0f, hi ? x2  : y1  };  // k 0..3
    const v2f a1 = { hi ? x3 : m4 , hi ? m7  : z2  };   // k 4..7
    const v2f a2 = { hi ? x4 : m8 , hi ? m11 : z3  };   // k 8..11
    const v2f a3 = { hi ? z4 : m12, hi ? 0.f : m13 };   // k 12..15 (15 = pad)

    v8f acc = {0.f,0.f,0.f,0.f,0.f,0.f,0.f,0.f};
    acc = __builtin_amdgcn_wmma_f32_16x16x4_f32(false, a0, false, b0, (short)0, acc, false, false);
    acc = __builtin_amdgcn_wmma_f32_16x16x4_f32(false, a1, false, b1, (short)0, acc, false, false);
    acc = __builtin_amdgcn_wmma_f32_16x16x4_f32(false, a2, false, b2, (short)0, acc, false, false);
    acc = __builtin_amdgcn_wmma_f32_16x16x4_f32(false, a3, false, b3, (short)0, acc, false, false);
    return acc;
}

__launch_bounds__(256, 4)
__global__ void hnn_grad_wmma(const float* __restrict__ y,
                              const float* __restrict__ C,
                              float* __restrict__ out,
                              int npts, int ntiles)
{
    const int  lane   = (int)(threadIdx.x & 31u);
    const int  col    = lane & 15;          // N-column this lane owns in B/C/D
    const bool hi     = (lane & 16) != 0;   // K-pair {2,3} half of A/B operands
    const bool storer = col < 2;

    // ---- B operand (coefficients), invariant across tiles; C[] loads are uniform ----
    // Monomial order k=0..14: (0,0)(1,0)(0,1)(2,0)(1,1)(0,2)(3,0)(2,1)(1,2)(0,3)
    //                         (4,0)(3,1)(2,2)(1,3)(0,4); k=15 pad.
    const float w0  = col==0 ? C[2] *1.0f : (col==1 ? -C[1] *1.0f : 0.0f);
    const float w1  = col==0 ? C[4] *1.0f : (col==1 ? -C[3] *2.0f : 0.0f);
    const float w2  = col==0 ? C[5] *2.0f : (col==1 ? -C[4] *1.0f : 0.0f);
    const float w3  = col==0 ? C[7] *1.0f : (col==1 ? -C[6] *3.0f : 0.0f);
    const float w4  = col==0 ? C[8] *2.0f : (col==1 ? -C[7] *2.0f : 0.0f);
    const float w5  = col==0 ? C[9] *3.0f : (col==1 ? -C[8] *1.0f : 0.0f);
    const float w6  = col==0 ? C[11]*1.0f : (col==1 ? -C[10]*4.0f : 0.0f);
    const float w7  = col==0 ? C[12]*2.0f : (col==1 ? -C[11]*3.0f : 0.0f);
    const float w8  = col==0 ? C[13]*3.0f : (col==1 ? -C[12]*2.0f : 0.0f);
    const float w9  = col==0 ? C[14]*4.0f : (col==1 ? -C[13]*1.0f : 0.0f);
    const float w10 = col==0 ? C[16]*1.0f : (col==1 ? -C[15]*5.0f : 0.0f);
    const float w11 = col==0 ? C[17]*2.0f : (col==1 ? -C[16]*4.0f : 0.0f);
    const float w12 = col==0 ? C[18]*3.0f : (col==1 ? -C[17]*3.0f : 0.0f);
    const float w13 = col==0 ? C[19]*4.0f : (col==1 ? -C[18]*2.0f : 0.0f);
    const float w14 = col==0 ? C[20]*5.0f : (col==1 ? -C[19]*1.0f : 0.0f);
    const float w15 = 0.0f;

    // B 4x16 chunks: VGPR0 = rows {K0 | K2}, VGPR1 = rows {K1 | K3} across lane halves.
    const v2f b0 = { hi ? w2  : w0 , hi ? w3  : w1  };
    const v2f b1 = { hi ? w6  : w4 , hi ? w7  : w5  };
    const v2f b2 = { hi ? w10 : w8 , hi ? w11 : w9  };
    const v2f b3 = { hi ? w14 : w12, hi ? w15 : w13 };

    const int wave   = (int)((blockIdx.x * blockDim.x + threadIdx.x) >> 5);
    const int nwaves = (int)((gridDim.x * blockDim.x) >> 5);

    const int nfull  = npts >> 4;    // tiles with no bounds hazard
    const int npairs = nfull >> 1;   // bulk: 2 tiles (32 points, 256B) per iteration

    for (int pr = wave; pr < npairs; pr += nwaves) {
        const int tile = pr * 2;
        const float* yb = y + (size_t)tile * 32 + 2 * col;   // this lane's point, tile0

        // Phase 1: issue both 8B loads back-to-back (tile1 at +128B imm offset).
        const v2f yv0 = *reinterpret_cast<const v2f*>(yb);
        const v2f yv1 = *reinterpret_cast<const v2f*>(yb + 32);

        // Stream-ahead prefetch of this wave's next 256B chunk.
        __builtin_prefetch(y + (size_t)(pr + nwaves) * 64 + 2 * col, 0, 0);
        __builtin_prefetch(y + (size_t)(pr + nwaves) * 64 + 32 + 2 * col, 0, 0);

        // Phase 2: two independent WMMA accumulation chains (interleave in the pipe).
        const v8f acc0 = hnn_compute(yv0, hi, b0, b1, b2, b3);
        const v8f acc1 = hnn_compute(yv1, hi, b0, b1, b2, b3);

        // Phase 3: one exec region, 16 stores in a single clause.
        // D layout: VGPR v -> (M = v + 8*hi, N = col); tile1 rows at +32 floats.
        if (storer) {
            float* o = out + (size_t)(tile * 16 + (hi ? 8 : 0)) * 2 + col;
            #pragma unroll
            for (int v = 0; v < 8; ++v) o[2 * v]      = acc0[v];  // imm 0..56B*? (8B stride)
            #pragma unroll
            for (int v = 0; v < 8; ++v) o[32 + 2 * v] = acc1[v];  // imm 128..184B
        }
    }

    // Leftovers: at most one full tile (nfull odd) + one partial tile.
    for (int tile = npairs * 2 + wave; tile < ntiles; tile += nwaves) {
        int p  = tile * 16 + col;
        int pc = (p < npts) ? p : (npts - 1);
        const v2f yv = *reinterpret_cast<const v2f*>(y + 2u * (unsigned)pc);
        const v8f acc = hnn_compute(yv, hi, b0, b1, b2, b3);
        if (storer) {
            const int rowbase = tile * 16 + (hi ? 8 : 0);
            float* o = out + (size_t)rowbase * 2 + col;
            #pragma unroll
            for (int v = 0; v < 8; ++v)
                if ((rowbase + v) < npts) o[2 * v] = acc[v];
        }
    }
}

extern "C" void kernel_launch(void* const* d_in, const int* in_sizes, int n_in,
                              void* d_out, int out_size, void* d_ws, size_t ws_size,
                              hipStream_t stream) {
    // inputs: 0:t[1], 1:y[N*2], 2:C[21], 3:L[4]  (t unused; L is the fixed
    // symplectic rotation, baked into the coefficient matrix W)
    const float* y  = (const float*)d_in[1];
    const float* Cc = (const float*)d_in[2];
    float* out      = (float*)d_out;

    const int npts   = in_sizes[1] / 2;
    const int ntiles = (npts + 15) / 16;

    const int threads       = 256;               // 8 wave32 per block
    const int wavesPerBlock = threads / 32;
    int blocks = (ntiles + wavesPerBlock - 1) / wavesPerBlock;
    if (blocks > 2048) blocks = 2048;            // persistent waves
    if (blocks < 1)    blocks = 1;

    hnn_grad_wmma<<<blocks, threads, 0, stream>>>(y, Cc, out, npts, ntiles);
}